// GATModel_22789096472973
// MI455X (gfx1250) — compile-verified
//
#include <hip/hip_runtime.h>
#include <hip/hip_bf16.h>
#include <math.h>

typedef __attribute__((ext_vector_type(2))) float v2f;
typedef __attribute__((ext_vector_type(8))) float v8f;

// ---------------------------------------------------------------------------
// helpers
// ---------------------------------------------------------------------------
__device__ __forceinline__ float wave_sum32(float v) {
#pragma unroll
  for (int off = 16; off > 0; off >>= 1) v += __shfl_xor(v, off, 32);
  return v;
}

// float atomic max via sign-split integer atomics (bit order of IEEE floats)
__device__ __forceinline__ void atomicMaxF(float* addr, float val) {
  if (val >= 0.0f) atomicMax((int*)addr, __float_as_int(val));
  else             atomicMin((unsigned int*)addr, __float_as_uint(val));
}

__global__ void fill_f32(float* __restrict__ p, float v, int n) {
  int i = blockIdx.x * blockDim.x + threadIdx.x;
  if (i < n) p[i] = v;
}

// ---------------------------------------------------------------------------
// C[N,H] = X[N,K] @ W[K,H] + bias    -- fp32 WMMA 16x16x4, one 16x16 tile/wave
// ---------------------------------------------------------------------------
template <int K, int H>
__global__ void gemm_bias_wmma(const float* __restrict__ X, const float* __restrict__ W,
                               const float* __restrict__ bias, float* __restrict__ C, int N) {
  constexpr int COLT = H / 16;      // column tiles (4 for H=64, 2 for H=32)
  constexpr int ROWT = 8 / COLT;    // row tiles per 8-wave block
  const int wave = threadIdx.x >> 5;
  const int lane = threadIdx.x & 31;
  const int rt = wave / COLT, ct = wave % COLT;
  const int row0 = (blockIdx.x * ROWT + rt) * 16;
  if (row0 >= N) return;
  const int col0 = ct * 16;
  const int half = lane >> 4, l16 = lane & 15;

  const float* __restrict__ arow = X + (size_t)(row0 + l16) * K;
  v8f acc = {};
#pragma unroll 4
  for (int k = 0; k < K; k += 4) {
    const int ka = k + 2 * half;
    v2f a, b;
    // A 16x4: lane m=l16; VGPR0=K(2*half), VGPR1=K(2*half+1)  -> contiguous pair
    a.x = arow[ka];
    a.y = arow[ka + 1];
    // B 4x16: lane n=l16; VGPR0 holds row K=2*half, VGPR1 row K=2*half+1
    b.x = W[(size_t)ka * H + col0 + l16];
    b.y = W[(size_t)(ka + 1) * H + col0 + l16];
    acc = __builtin_amdgcn_wmma_f32_16x16x4_f32(false, a, false, b, (short)0, acc,
                                                false, false);
  }
  const float bv = bias[col0 + l16];
  // C/D: VGPR v -> row (v + 8*half), col l16
  float* __restrict__ crow = C + (size_t)(row0 + 8 * half) * H + col0 + l16;
#pragma unroll
  for (int v = 0; v < 8; ++v) crow[(size_t)v * H] = acc[v] + bv;
}

// ---------------------------------------------------------------------------
// Edge pass 1: e = att . leaky_relu(xl[src]+xr[dst], 0.2); segment max into m
// one wave per edge; edges >= E are self loops
// ---------------------------------------------------------------------------
template <int H>
__global__ void edge_scores(const float* __restrict__ xl, const float* __restrict__ xr,
                            const float* __restrict__ att, const int* __restrict__ ei,
                            int E, int Etot, float* __restrict__ ebuf,
                            float* __restrict__ m) {
  const int gid = blockIdx.x * blockDim.x + threadIdx.x;
  const int e = gid >> 5, lane = gid & 31;
  if (e >= Etot) return;
  int s, d;
  if (e < E) { s = ei[e]; d = ei[E + e]; } else { s = d = e - E; }

  float partial;
  if (H == 64) {
    const int c0 = lane * 2;
    const float2 a = *(const float2*)(xl + (size_t)s * 64 + c0);
    const float2 b = *(const float2*)(xr + (size_t)d * 64 + c0);
    float h0 = a.x + b.x; h0 = h0 > 0.0f ? h0 : 0.2f * h0;
    float h1 = a.y + b.y; h1 = h1 > 0.0f ? h1 : 0.2f * h1;
    partial = h0 * att[c0] + h1 * att[c0 + 1];
  } else {
    float hv = xl[(size_t)s * H + lane] + xr[(size_t)d * H + lane];
    hv = hv > 0.0f ? hv : 0.2f * hv;
    partial = hv * att[lane];
  }
  const float tot = wave_sum32(partial);
  if (lane == 0) {
    ebuf[e] = tot;
    atomicMaxF(m + d, tot);
  }
}

// ---------------------------------------------------------------------------
// Edge pass 2: ex = exp(e - m[dst]); denom[dst]+=ex; acc[dst] += ex*xl[src]
// (normalization by denom folded into finalize -> identical to alpha softmax)
// ---------------------------------------------------------------------------
template <int H>
__global__ void edge_aggregate(const float* __restrict__ xl, const int* __restrict__ ei,
                               int E, int Etot, const float* __restrict__ ebuf,
                               const float* __restrict__ m, float* __restrict__ denom,
                               float* __restrict__ acc) {
  const int gid = blockIdx.x * blockDim.x + threadIdx.x;
  const int e = gid >> 5, lane = gid & 31;
  if (e >= Etot) return;
  int s, d;
  if (e < E) { s = ei[e]; d = ei[E + e]; } else { s = d = e - E; }

  const float ex = expf(ebuf[e] - m[d]);
  if (lane == 0) atomicAdd(denom + d, ex);
  if (H == 64) {
    const int c0 = lane * 2;
    const float2 a = *(const float2*)(xl + (size_t)s * 64 + c0);
    atomicAdd(acc + (size_t)d * 64 + c0, ex * a.x);
    atomicAdd(acc + (size_t)d * 64 + c0 + 1, ex * a.y);
  } else {
    atomicAdd(acc + (size_t)d * H + lane, ex * xl[(size_t)s * H + lane]);
  }
}

// ---------------------------------------------------------------------------
// finalize layer1: h = leaky_relu(acc/denom + bias, 0.01)   (in place OK)
// ---------------------------------------------------------------------------
__global__ void finalize_layer1(const float* __restrict__ acc, const float* __restrict__ denom,
                                const float* __restrict__ bias, float* __restrict__ h, int N) {
  const int i = blockIdx.x * blockDim.x + threadIdx.x;
  if (i >= N * 64) return;
  const int n = i >> 6, c = i & 63;
  float v = acc[i] / (denom[n] + 1e-16f) + bias[c];
  h[i] = v > 0.0f ? v : 0.01f * v;
}

// ---------------------------------------------------------------------------
// finalize layer2 + final linear: out[n] = (acc/denom + bias) . Wlin + blin
// one wave per node (H=32 -> one lane per channel)
// ---------------------------------------------------------------------------
__global__ void finalize_layer2(const float* __restrict__ acc, const float* __restrict__ denom,
                                const float* __restrict__ bias, const float* __restrict__ Wlin,
                                const float* __restrict__ blin, float* __restrict__ out, int N) {
  const int gid = blockIdx.x * blockDim.x + threadIdx.x;
  const int n = gid >> 5, lane = gid & 31;
  if (n >= N) return;
  const float v = acc[(size_t)n * 32 + lane] / (denom[n] + 1e-16f) + bias[lane];
  const float tot = wave_sum32(v * Wlin[lane]);
  if (lane == 0) out[n] = tot + blin[0];
}

// ---------------------------------------------------------------------------
static inline int cdiv(int a, int b) { return (a + b - 1) / b; }

extern "C" void kernel_launch(void* const* d_in, const int* in_sizes, int n_in,
                              void* d_out, int out_size, void* d_ws, size_t ws_size,
                              hipStream_t stream) {
  const float* x     = (const float*)d_in[0];
  const int*   ei    = (const int*)  d_in[1];
  const float* W1l   = (const float*)d_in[2];
  const float* b1l   = (const float*)d_in[3];
  const float* W1r   = (const float*)d_in[4];
  const float* b1r   = (const float*)d_in[5];
  const float* att1  = (const float*)d_in[6];
  const float* bias1 = (const float*)d_in[7];
  const float* W2l   = (const float*)d_in[8];
  const float* b2l   = (const float*)d_in[9];
  const float* W2r   = (const float*)d_in[10];
  const float* b2r   = (const float*)d_in[11];
  const float* att2  = (const float*)d_in[12];
  const float* bias2 = (const float*)d_in[13];
  const float* Wlin  = (const float*)d_in[14];
  const float* blin  = (const float*)d_in[15];
  float* out = (float*)d_out;

  const int N    = in_sizes[0] / 128;
  const int E    = in_sizes[1] / 2;
  const int Etot = E + N;

  float* ws = (float*)d_ws;
  size_t off = 0;
  float* xl1  = ws + off; off += (size_t)N * 64;
  float* xr1  = ws + off; off += (size_t)N * 64;
  float* acc1 = ws + off; off += (size_t)N * 64;   // accumulator, then h (in place)
  float* xl2  = ws + off; off += (size_t)N * 32;
  float* xr2  = ws + off; off += (size_t)N * 32;
  float* acc2 = ws + off; off += (size_t)N * 32;
  float* ebuf = ws + off; off += (size_t)Etot;
  float* mbuf = ws + off; off += (size_t)N;
  float* dbuf = ws + off; off += (size_t)N;
  (void)ws_size; (void)n_in; (void)out_size;

  const int TB = 256;                 // 8 waves per block (wave32)
  const int rowTiles = cdiv(N, 16);
  const int eblocks = cdiv(Etot, TB / 32);

  // ---------------- layer 1 ----------------
  gemm_bias_wmma<128, 64><<<cdiv(rowTiles, 2), TB, 0, stream>>>(x, W1l, b1l, xl1, N);
  gemm_bias_wmma<128, 64><<<cdiv(rowTiles, 2), TB, 0, stream>>>(x, W1r, b1r, xr1, N);

  fill_f32<<<cdiv(N * 64, TB), TB, 0, stream>>>(acc1, 0.0f, N * 64);
  fill_f32<<<cdiv(N, TB), TB, 0, stream>>>(mbuf, -INFINITY, N);
  fill_f32<<<cdiv(N, TB), TB, 0, stream>>>(dbuf, 0.0f, N);

  edge_scores<64><<<eblocks, TB, 0, stream>>>(xl1, xr1, att1, ei, E, Etot, ebuf, mbuf);
  edge_aggregate<64><<<eblocks, TB, 0, stream>>>(xl1, ei, E, Etot, ebuf, mbuf, dbuf, acc1);
  finalize_layer1<<<cdiv(N * 64, TB), TB, 0, stream>>>(acc1, dbuf, bias1, acc1, N);

  // ---------------- layer 2 ----------------
  gemm_bias_wmma<64, 32><<<cdiv(rowTiles, 4), TB, 0, stream>>>(acc1, W2l, b2l, xl2, N);
  gemm_bias_wmma<64, 32><<<cdiv(rowTiles, 4), TB, 0, stream>>>(acc1, W2r, b2r, xr2, N);

  fill_f32<<<cdiv(N * 32, TB), TB, 0, stream>>>(acc2, 0.0f, N * 32);
  fill_f32<<<cdiv(N, TB), TB, 0, stream>>>(mbuf, -INFINITY, N);
  fill_f32<<<cdiv(N, TB), TB, 0, stream>>>(dbuf, 0.0f, N);

  edge_scores<32><<<eblocks, TB, 0, stream>>>(xl2, xr2, att2, ei, E, Etot, ebuf, mbuf);
  edge_aggregate<32><<<eblocks, TB, 0, stream>>>(xl2, ei, E, Etot, ebuf, mbuf, dbuf, acc2);

  finalize_layer2<<<cdiv(N, TB / 32), TB, 0, stream>>>(acc2, dbuf, bias2, Wlin, blin, out, N);
}